// SpatialLowRankConv2d_14611478741671
// MI455X (gfx1250) — compile-verified
//
#include <hip/hip_runtime.h>
#include <hip/hip_bf16.h>

// Effective computation: dense 3x3 conv (64 -> 64 ch) with
// Keff[co,ci,kh,kw] = sum_r h_weight[g,kh] * v_weight[g,kw], g = ci*256 + co*4 + r.
// Implemented as implicit GEMM on V_WMMA_F32_16X16X4_F32 (full fp32 precision).

typedef __attribute__((ext_vector_type(2))) float v2f;
typedef __attribute__((ext_vector_type(8))) float v8f;

#define CIN   64
#define COUT  64
#define RANK  4
#define HH    56
#define WW    56
#define KDIM  (CIN * 9)        // 576 : GEMM K, ordered (kh*3+kw)*64 + ci
#define TILE_W 58              // cols -1..56 (halo)
#define CI_STRIDE (3 * TILE_W) // 174 floats per ci in LDS tile

// ---------------------------------------------------------------------------
// Kernel 1: collapse the low-rank separable weights into a dense 3x3 kernel.
// Keff layout: [co][ (kh*3+kw)*64 + ci ]  (so A-fragment float2 loads are
// contiguous along K).
// ---------------------------------------------------------------------------
__global__ void prep_keff(const float* __restrict__ vw,   // [G,1,1,3]
                          const float* __restrict__ hw,   // [G,1,3,1]
                          float* __restrict__ keff) {
    int idx = blockIdx.x * blockDim.x + threadIdx.x;
    if (idx >= COUT * KDIM) return;
    int co = idx / KDIM;
    int k  = idx - co * KDIM;
    int kk = k >> 6;           // kh*3+kw  (0..8)
    int ci = k & 63;
    int kh = kk / 3;
    int kw = kk - kh * 3;
    int gbase = ci * (COUT * RANK) + co * RANK;  // + r
    float s = 0.f;
#pragma unroll
    for (int r = 0; r < RANK; ++r) {
        int g = gbase + r;
        s += hw[g * 3 + kh] * vw[g * 3 + kw];
    }
    keff[idx] = s;
}

// ---------------------------------------------------------------------------
// Kernel 2: implicit-GEMM conv via WMMA f32 16x16x4.
// Grid: one block per (b, h) output row.  512 threads = 16 waves
//   wave -> (co_tile = wave>>2, w_tile = wave&3), each owns a 16(co) x 16(w)
//   output tile and runs K = 576 in 144 chained WMMA steps.
// LDS: x slab [ci=64][ry=0..2][col=0..57] with zero halo.
// ---------------------------------------------------------------------------
__global__ __launch_bounds__(512)
void conv_wmma(const float* __restrict__ x,     // [B,64,56,56]
               const float* __restrict__ keff,  // [64][576]
               const float* __restrict__ bias,  // [64]
               float* __restrict__ out) {       // [B,64,56,56]
    __shared__ float tile[CIN * CI_STRIDE];     // 44544 bytes

    const int bh = blockIdx.x;        // 0 .. B*H-1
    const int b  = bh / HH;
    const int h  = bh - b * HH;
    const int tid = threadIdx.x;

    // ---- cooperative stage of the 3-row input slab (zero-padded halo) ----
    const float* xb = x + (size_t)b * CIN * HH * WW;
    for (int e = tid; e < CIN * CI_STRIDE; e += 512) {
        int ci  = e / CI_STRIDE;
        int rem = e - ci * CI_STRIDE;
        int ry  = rem / TILE_W;       // 0..2  -> input row h+ry-1
        int col = rem - ry * TILE_W;  // 0..57 -> input col col-1
        int y  = h + ry - 1;
        int wc = col - 1;
        float v = 0.f;
        if ((unsigned)y < (unsigned)HH && (unsigned)wc < (unsigned)WW)
            v = xb[(ci * HH + y) * WW + wc];
        tile[e] = v;
    }
    __syncthreads();

    const int wave  = tid >> 5;       // 0..15 (wave32)
    const int lane  = tid & 31;
    const int half  = lane >> 4;      // 0: lanes 0-15, 1: lanes 16-31
    const int laneW = lane & 15;
    const int co0 = (wave >> 2) * 16; // co tile base
    const int w0  = (wave & 3) * 16;  // w  tile base (w0=48 tile is masked)

    // A fragment source: lane holds A[M=laneW][K = k0 + 2*half + {0,1}]
    const float* aRow = keff + (co0 + laneW) * KDIM;

    v8f acc = {0.f, 0.f, 0.f, 0.f, 0.f, 0.f, 0.f, 0.f};

    for (int kk = 0; kk < 9; ++kk) {
        const int kh = kk / 3;
        const int kw = kk - kh * 3;
        // B element for (ci, n=w0+laneW): x[ci][h+kh-1][w0+laneW+kw-1]
        //   -> lds col = w0+laneW+kw (halo shift +1)
        const float* bBase = tile + 2 * half * CI_STRIDE
                                  + kh * TILE_W + w0 + laneW + kw;
        const float* aK = aRow + kk * 64 + 2 * half;
#pragma unroll
        for (int t = 0; t < 16; ++t) {           // ci group = 4t .. 4t+3
            v2f a = *(const v2f*)(aK + 4 * t);   // K = 4t+2h, 4t+2h+1
            v2f bv;
            bv.x = bBase[(4 * t)     * CI_STRIDE]; // ci = 4t+2h
            bv.y = bBase[(4 * t + 1) * CI_STRIDE]; // ci = 4t+2h+1
            acc = __builtin_amdgcn_wmma_f32_16x16x4_f32(
                false, a, false, bv, (short)0, acc, false, false);
        }
    }

    // ---- store: acc[j] = out[co0 + 8*half + j][w0 + laneW] ----
    const int wN = w0 + laneW;
    if (wN < WW) {
        const int coB = co0 + half * 8;
        float* op = out + (((size_t)b * COUT + coB) * HH + h) * WW + wN;
#pragma unroll
        for (int j = 0; j < 8; ++j)
            op[(size_t)j * HH * WW] = acc[j] + bias[coB + j];
    }
}

extern "C" void kernel_launch(void* const* d_in, const int* in_sizes, int n_in,
                              void* d_out, int out_size, void* d_ws, size_t ws_size,
                              hipStream_t stream) {
    const float* x    = (const float*)d_in[0];  // [4,64,56,56]
    const float* vw   = (const float*)d_in[1];  // [16384,1,1,3]
    const float* hw   = (const float*)d_in[2];  // [16384,1,3,1]
    const float* bias = (const float*)d_in[3];  // [64]
    float* out  = (float*)d_out;                // [4,64,56,56]
    float* keff = (float*)d_ws;                 // 64*576 floats = 147456 B

    const int nK = COUT * KDIM;                 // 36864
    prep_keff<<<(nK + 255) / 256, 256, 0, stream>>>(vw, hw, keff);

    const int B = 4;
    conv_wmma<<<B * HH, 512, 0, stream>>>(x, keff, bias, out);
}